// MaxBlurPooling2D_73658689126583
// MI455X (gfx1250) — compile-verified
//
#include <hip/hip_runtime.h>

// ---------------------------------------------------------------------------
// Problem constants (from reference: x is (32,112,112,128) f32 NHWC).
// ---------------------------------------------------------------------------
static constexpr int NB   = 32;
static constexpr int HH   = 112;
static constexpr int WW   = 112;
static constexpr int CC   = 128;        // channels, innermost
static constexpr int C4   = CC / 4;     // 32 float4 per pixel == one wave32
static constexpr int HO   = 56;
static constexpr int WOUT = 56;

// Block tile: 2 (ho) x 4 (wo) output pixels, one wave32 per output pixel.
static constexpr int TDH = 2;
static constexpr int TDW = 4;
static constexpr int TR  = 2 * TDH + 3;          // 7 input rows in tile
static constexpr int TC  = 2 * TDW + 3;          // 11 input cols in tile
static constexpr int TELEM = TR * TC * C4;       // 2464 float4 elems in tile
static constexpr int THREADS = 256;              // 8 waves

static constexpr int GBH = HO / TDH;             // 28 tile rows
static constexpr int GBW = WOUT / TDW;           // 14 tile cols
static constexpr int BLOCKS = NB * GBH * GBW;    // 12544

typedef float v4f __attribute__((ext_vector_type(4)));

__device__ __forceinline__ v4f f4max(v4f a, v4f b) {
    v4f r;
    r.x = fmaxf(a.x, b.x);
    r.y = fmaxf(a.y, b.y);
    r.z = fmaxf(a.z, b.z);
    r.w = fmaxf(a.w, b.w);
    return r;
}

__device__ __forceinline__ void f4fma(v4f& acc, float s, v4f v) {
    acc.x = fmaf(s, v.x, acc.x);
    acc.y = fmaf(s, v.y, acc.y);
    acc.z = fmaf(s, v.z, acc.z);
    acc.w = fmaf(s, v.w, acc.w);
}

__device__ __forceinline__ int iclamp(int v, int lo, int hi) {
    return v < lo ? lo : (v > hi ? hi : v);
}

// Fused maxpool(2x2,s1,SAME-after) -> depthwise 3x3 blur(same) -> avgpool(2x2,s2).
// Blur+avg fold into one 4x4 stride-2 depthwise conv over the maxpooled map m:
//   out[ho,wo]  = sum_{i,j=0..3} m[2ho-1+i, 2wo-1+j] * K4[i][j]
//   K4[i][j]    = 0.25 * sum_{p,q in {0,1}, 0<=i-p<3, 0<=j-q<3} blur[i-p][j-q]
// Zero padding of m (conv) -> zero the border K4 weights per output pixel.
// -inf padding inside the maxpool -> clamp input indices (max over a
// duplicated element == max over the valid subset).
//
// The 7x11-pixel input tile (all 128 channels) is staged into LDS with
// gfx1250 async copies (global_load_async_to_lds_b128, ASYNCcnt-tracked),
// then each wave reads its 5x5 patch from LDS (ds_load_b128).
__global__ __launch_bounds__(THREADS) void mbp_fused_lds(const float* __restrict__ x,
                                                         const float* __restrict__ bk,
                                                         float* __restrict__ out) {
    __shared__ v4f tile[TELEM];          // 38.5 KB

    const int lane = threadIdx.x & 31;
    const int wv   = threadIdx.x >> 5;

    // Tile coordinates.
    int blk = blockIdx.x;
    const int bw = blk % GBW;            // tile col index
    blk /= GBW;
    const int bh = blk % GBH;            // tile row index
    const int b  = blk / GBH;            // batch
    const int ho0 = bh * TDH;
    const int wo0 = bw * TDW;

    // ---------------- stage input tile into LDS (async DMA) ----------------
    // Element e -> (pixel = e/32, lane-float4 = e%32). Threads are mapped so
    // each wave's 32 lanes cover one pixel-vector: one contiguous 512B burst.
    const unsigned lds_base = (unsigned)(uintptr_t)(&tile[0]);
#pragma unroll 1
    for (int e = threadIdx.x; e < TELEM; e += THREADS) {
        const int l   = e & 31;
        const int pix = e >> 5;
        const int ti  = pix / TC;
        const int tj  = pix - ti * TC;
        const int r   = iclamp(2 * ho0 - 1 + ti, 0, HH - 1);
        const int c   = iclamp(2 * wo0 - 1 + tj, 0, WW - 1);
        const unsigned goff = (unsigned)(((b * HH + r) * WW + c) * CC + l * 4) * 4u;
        const unsigned loff = lds_base + (unsigned)e * 16u;
        asm volatile("global_load_async_to_lds_b128 %0, %1, %2"
                     :
                     : "v"(loff), "v"(goff), "s"(x)
                     : "memory");
    }
    asm volatile("s_wait_asynccnt 0x0" ::: "memory");
    __syncthreads();

    // ---------------- per-wave output pixel ----------------
    const int dh = wv >> 2;              // 0..1
    const int dw = wv & 3;               // 0..3
    const int ho = ho0 + dh;
    const int wo = wo0 + dw;

    // Combined 4x4 kernel from the 3x3 blur (generic, read from d_in[1]).
    float k3[3][3];
#pragma unroll
    for (int u = 0; u < 3; ++u)
#pragma unroll
        for (int v = 0; v < 3; ++v)
            k3[u][v] = bk[u * 3 + v];

    const bool rv[4] = { ho > 0, true, true, ho < HO - 1 };
    const bool cv[4] = { wo > 0, true, true, wo < WOUT - 1 };

    float K4[4][4];
#pragma unroll
    for (int i = 0; i < 4; ++i) {
#pragma unroll
        for (int j = 0; j < 4; ++j) {
            float s = 0.0f;
#pragma unroll
            for (int pp = 0; pp < 2; ++pp) {
#pragma unroll
                for (int qq = 0; qq < 2; ++qq) {
                    const int u = i - pp, v = j - qq;
                    if (u >= 0 && u < 3 && v >= 0 && v < 3) s += k3[u][v];
                }
            }
            K4[i][j] = (rv[i] && cv[j]) ? 0.25f * s : 0.0f;
        }
    }

    // 5x5 patch for this wave inside the LDS tile.
    v4f acc = (v4f)0.0f;
    v4f ph[4];
#pragma unroll
    for (int i = 0; i < 5; ++i) {
        const int ti  = 2 * dh + i;
        const int pb  = ti * TC + 2 * dw;      // first pixel of this patch row

        v4f xr[5];
#pragma unroll
        for (int j = 0; j < 5; ++j)
            xr[j] = tile[(pb + j) * C4 + lane];

        v4f ch[4];
#pragma unroll
        for (int j = 0; j < 4; ++j)
            ch[j] = f4max(xr[j], xr[j + 1]);

        if (i > 0) {
            const int im = i - 1;
#pragma unroll
            for (int j = 0; j < 4; ++j) {
                v4f m = f4max(ph[j], ch[j]);   // 2x2 maxpool value
                f4fma(acc, K4[im][j], m);      // weight==0 masks padding
            }
        }
#pragma unroll
        for (int j = 0; j < 4; ++j) ph[j] = ch[j];
    }

    // Output is streaming (no reuse): non-temporal store keeps L2 for x.
    const int p = (b * HO + ho) * WOUT + wo;
    v4f* o4 = (v4f*)out;
    __builtin_nontemporal_store(acc, &o4[(size_t)p * C4 + lane]);
}

extern "C" void kernel_launch(void* const* d_in, const int* in_sizes, int n_in,
                              void* d_out, int out_size, void* d_ws, size_t ws_size,
                              hipStream_t stream) {
    (void)in_sizes; (void)n_in; (void)out_size; (void)d_ws; (void)ws_size;
    const float* x  = (const float*)d_in[0];   // (32,112,112,128) f32
    const float* bk = (const float*)d_in[1];   // (3,3,1,1) f32
    float* out = (float*)d_out;                // (32,56,56,128) f32

    mbp_fused_lds<<<BLOCKS, THREADS, 0, stream>>>(x, bk, out);
}